// RegionProposalNetwork_91268055040556
// MI455X (gfx1250) — compile-verified
//
#include <hip/hip_runtime.h>
#include <hip/hip_bf16.h>
#include <math.h>

typedef __attribute__((ext_vector_type(2))) float v2f;
typedef __attribute__((ext_vector_type(8))) float v8f;

#define NEG_INF (-__builtin_huge_valf())
#define C_IN   512
#define NPROP  36864          // 64*64*9
#define PRE_TOPK 6000
#define POST_K   300
#define LOG_MAX_F 4.135166556742356f   // log(1000/16)

// ---------------------------------------------------------------------------
// Kernel 1: 3x3 conv (batch 0) + bias + ReLU, implicit GEMM on f32 WMMA.
// Block = 256 thr (8 waves). Block tile: M=128 (2 image rows) x N=64.
// Wave tile: 32x32 = 2x2 WMMA 16x16 tiles. K loop: 32 chunks of 16 ch x 9 taps.
// LDS layouts are channel-innermost so each WMMA operand's two K values are
// one aligned ds_load_b64 (no VGPR-pair marshalling moves).
// ---------------------------------------------------------------------------
__global__ __launch_bounds__(256)
void rpn_conv3x3_wmma(const float* __restrict__ feat0,
                      const float* __restrict__ wOIHW,
                      const float* __restrict__ bias,
                      float* __restrict__ t0)
{
    __shared__ float sA[4 * 66 * 16];   // [halo_row][halo_col][ch]
    __shared__ float sB[9 * 64 * 16];   // [tap][n][ch]

    const int tid  = threadIdx.x;
    const int lane = tid & 31;
    const int wid  = tid >> 5;
    const int wm   = (wid & 3) * 32;    // wave M offset in block tile
    const int wn   = (wid >> 2) * 32;   // wave N offset in block tile

    const int mBase = blockIdx.x * 128;         // 2 consecutive image rows
    const int y0    = mBase >> 6;
    const int nBase = blockIdx.y * 64;

    v8f acc00 = {}; v8f acc01 = {}; v8f acc10 = {}; v8f acc11 = {};

    const int laneM = lane & 15;
    const int laneN = lane & 15;
    const int kh    = lane >> 4;        // K half: lanes 16-31 hold K=2,3
    const int ml0 = wm + laneM;         // local m for A tile 0
    const int ml1 = ml0 + 16;           // local m for A tile 1
    const int ly0 = ml0 >> 6, lx0 = ml0 & 63;
    const int ly1 = ml1 >> 6, lx1 = ml1 & 63;

    for (int c0 = 0; c0 < C_IN; c0 += 16) {
        // --- stage A halo tile: 4 rows x 66 cols x 16 ch (zero padded) ---
        // idx = c*264 + r*66 + col keeps global reads coalesced in col.
        for (int idx = tid; idx < 16 * 4 * 66; idx += 256) {
            int c   = idx / 264;
            int rem = idx - c * 264;
            int r   = rem / 66;
            int col = rem - r * 66;
            int gy = y0 - 1 + r;
            int gx = col - 1;
            float v = 0.0f;
            if ((unsigned)gy < 64u && (unsigned)gx < 64u)
                v = feat0[(((c0 + c) * 64) + gy) * 64 + gx];
            sA[(r * 66 + col) * 16 + c] = v;
        }
        // --- stage B weights: 9 taps x 64 n x 16 ch ---
        // idx = (n*16 + c)*9 + t  ==> global address is contiguous in idx.
        for (int idx = tid; idx < 9 * 16 * 64; idx += 256) {
            int n   = idx / 144;
            int rem = idx - n * 144;
            int c   = rem / 9;
            int t   = rem - c * 9;
            sB[(t * 64 + n) * 16 + c] =
                wOIHW[((nBase + n) * C_IN + (c0 + c)) * 9 + t];
        }
        if (c0 + 16 < C_IN)   // warm L2 for next weight chunk (global_prefetch_b8)
            __builtin_prefetch(&wOIHW[((nBase + (tid & 63)) * C_IN + (c0 + 16)) * 9], 0, 1);
        __syncthreads();

        for (int tap = 0; tap < 9; ++tap) {
            const int dy = tap / 3, dx = tap % 3;
            // per-lane base pointers; kk indexes as immediate ds offsets
            const v2f* A0 = (const v2f*)&sA[((ly0 + dy) * 66 + (lx0 + dx)) * 16 + 2 * kh];
            const v2f* A1 = (const v2f*)&sA[((ly1 + dy) * 66 + (lx1 + dx)) * 16 + 2 * kh];
            const v2f* B0 = (const v2f*)&sB[(tap * 64 + wn + laneN) * 16 + 2 * kh];
#pragma unroll
            for (int kk = 0; kk < 16; kk += 4) {
                const int o = kk >> 1;           // v2f element offset
                v2f a0 = A0[o];
                v2f a1 = A1[o];
                v2f b0 = B0[o];
                v2f b1 = B0[o + 128];            // N+16 tile: +16*16 floats
                acc00 = __builtin_amdgcn_wmma_f32_16x16x4_f32(false, a0, false, b0, (short)0, acc00, false, false);
                acc01 = __builtin_amdgcn_wmma_f32_16x16x4_f32(false, a0, false, b1, (short)0, acc01, false, false);
                acc10 = __builtin_amdgcn_wmma_f32_16x16x4_f32(false, a1, false, b0, (short)0, acc10, false, false);
                acc11 = __builtin_amdgcn_wmma_f32_16x16x4_f32(false, a1, false, b1, (short)0, acc11, false, false);
            }
        }
        __syncthreads();
    }

    // --- epilogue: bias + ReLU, store t0[m][n] (C/D layout: vgpr v -> M=v+8*kh, N=lane&15) ---
    const int nc0 = nBase + wn + (lane & 15);
    const int nc1 = nc0 + 16;
    const float bv0 = bias[nc0];
    const float bv1 = bias[nc1];
    const int mrow0 = mBase + wm;
    const int mrow1 = mrow0 + 16;
#pragma unroll
    for (int v = 0; v < 8; ++v) {
        const int Mofs = v + 8 * kh;
        t0[(mrow0 + Mofs) * C_IN + nc0] = fmaxf(acc00[v] + bv0, 0.0f);
        t0[(mrow0 + Mofs) * C_IN + nc1] = fmaxf(acc01[v] + bv1, 0.0f);
        t0[(mrow1 + Mofs) * C_IN + nc0] = fmaxf(acc10[v] + bv0, 0.0f);
        t0[(mrow1 + Mofs) * C_IN + nc1] = fmaxf(acc11[v] + bv1, 0.0f);
    }
}

// ---------------------------------------------------------------------------
// Kernel 2: fused 1x1 heads + sigmoid + anchor decode + clip + validity.
// One thread per (spatial m, anchor a): 5 dot products of length 512.
// ---------------------------------------------------------------------------
__global__ __launch_bounds__(256)
void rpn_head_decode(const float* __restrict__ t0,
                     const float* __restrict__ clsw, const float* __restrict__ clsb,
                     const float* __restrict__ regw, const float* __restrict__ regb,
                     float* __restrict__ props, float* __restrict__ rawsc,
                     float* __restrict__ validf)
{
    const int i = blockIdx.x * blockDim.x + threadIdx.x;
    if (i >= NPROP) return;
    const int m = i / 9;
    const int a = i - m * 9;

    const float* tr = t0   + m * C_IN;
    const float* wc = clsw + a * C_IN;
    const float* w0 = regw + (a * 4 + 0) * C_IN;
    const float* w1 = regw + (a * 4 + 1) * C_IN;
    const float* w2 = regw + (a * 4 + 2) * C_IN;
    const float* w3 = regw + (a * 4 + 3) * C_IN;

    float sc = 0.f, d0 = 0.f, d1 = 0.f, d2 = 0.f, d3 = 0.f;
    for (int c = 0; c < C_IN; c += 4) {
        const float4 tv = *(const float4*)(tr + c);
        float4 v;
        v = *(const float4*)(wc + c); sc += tv.x*v.x + tv.y*v.y + tv.z*v.z + tv.w*v.w;
        v = *(const float4*)(w0 + c); d0 += tv.x*v.x + tv.y*v.y + tv.z*v.z + tv.w*v.w;
        v = *(const float4*)(w1 + c); d1 += tv.x*v.x + tv.y*v.y + tv.z*v.z + tv.w*v.w;
        v = *(const float4*)(w2 + c); d2 += tv.x*v.x + tv.y*v.y + tv.z*v.z + tv.w*v.w;
        v = *(const float4*)(w3 + c); d3 += tv.x*v.x + tv.y*v.y + tv.z*v.z + tv.w*v.w;
    }
    sc += clsb[a];
    d0 += regb[a * 4 + 0]; d1 += regb[a * 4 + 1];
    d2 += regb[a * 4 + 2]; d3 += regb[a * 4 + 3];

    const float score = 1.0f / (1.0f + expf(-sc));

    // anchors: stride 16, centers at +8; base sizes from SCALES x RATIOS
    const int   y = m >> 6, x = m & 63;
    const float cx = (float)x * 16.0f + 8.0f;
    const float cy = (float)y * 16.0f + 8.0f;
    const float scl[3] = {128.f, 256.f, 512.f};
    const float rat[3] = {0.5f, 1.0f, 2.0f};
    const float s = scl[a / 3];
    const float r = rat[a % 3];
    const float aw = s * sqrtf(1.0f / r);
    const float ah = s * sqrtf(r);

    const float pcx = d0 * aw + cx;
    const float pcy = d1 * ah + cy;
    const float pw  = expf(fminf(d2, LOG_MAX_F)) * aw;
    const float ph  = expf(fminf(d3, LOG_MAX_F)) * ah;

    float x1 = fminf(fmaxf(pcx - 0.5f * pw, 0.f), 1024.f);
    float y1 = fminf(fmaxf(pcy - 0.5f * ph, 0.f), 1024.f);
    float x2 = fminf(fmaxf(pcx + 0.5f * pw, 0.f), 1024.f);
    float y2 = fminf(fmaxf(pcy + 0.5f * ph, 0.f), 1024.f);
    const bool valid = ((x2 - x1) >= 1.0f) && ((y2 - y1) >= 1.0f);

    float4 p; p.x = x1; p.y = y1; p.z = x2; p.w = y2;
    *(float4*)(props + i * 4) = p;
    rawsc[i]  = score;
    validf[i] = valid ? 1.0f : 0.0f;
}

// ---------------------------------------------------------------------------
// Kernel 3: exact top-6000 cut (binary search on positive-float bit keys)
// + 300-iteration greedy NMS. Single persistent block of 1024 threads.
// ---------------------------------------------------------------------------
__global__ __launch_bounds__(1024)
void rpn_topk_nms(const float* __restrict__ props,
                  const float* __restrict__ rawsc,
                  const float* __restrict__ validf,
                  float* __restrict__ live,
                  float* __restrict__ areas,
                  float* __restrict__ out)
{
    __shared__ float sVal[1024];
    __shared__ int   sIdx[1024];
    const int tid = threadIdx.x;

    // ---- threshold T = 6000th-largest score key (scores > 0 => bit-monotone) ----
    unsigned T = 0u;
    for (int bit = 31; bit >= 0; --bit) {
        const unsigned cand = T | (1u << bit);
        int cnt = 0;
        for (int idx = tid; idx < NPROP; idx += 1024)
            cnt += (__float_as_uint(rawsc[idx]) >= cand) ? 1 : 0;
        sIdx[tid] = cnt;
        __syncthreads();
        for (int off = 512; off > 0; off >>= 1) {
            if (tid < off) sIdx[tid] += sIdx[tid + off];
            __syncthreads();
        }
        if (sIdx[0] >= PRE_TOPK) T = cand;
        __syncthreads();
    }

    // ---- init live scores and areas ----
    for (int idx = tid; idx < NPROP; idx += 1024) {
        const float4 p = *(const float4*)(props + idx * 4);
        areas[idx] = (p.z - p.x) * (p.w - p.y);
        const bool cand = (__float_as_uint(rawsc[idx]) >= T) && (validf[idx] > 0.5f);
        live[idx] = cand ? rawsc[idx] : NEG_INF;
    }
    __syncthreads();

    // ---- greedy NMS: 300 x (argmax + IoU suppression) ----
    for (int it = 0; it < POST_K; ++it) {
        float bv = NEG_INF;
        int   bi = NPROP;
        for (int idx = tid; idx < NPROP; idx += 1024) {
            const float v = live[idx];
            if (v > bv || (v == bv && idx < bi)) { bv = v; bi = idx; }
        }
        sVal[tid] = bv; sIdx[tid] = bi;
        __syncthreads();
        for (int off = 512; off > 0; off >>= 1) {
            if (tid < off) {
                const float ov = sVal[tid + off];
                const int   oi = sIdx[tid + off];
                if (ov > sVal[tid] || (ov == sVal[tid] && oi < sIdx[tid])) {
                    sVal[tid] = ov; sIdx[tid] = oi;
                }
            }
            __syncthreads();
        }
        const int   best  = sIdx[0];
        const float bestv = sVal[0];
        const bool  ok    = bestv > NEG_INF;
        const float4 bb   = *(const float4*)(props + best * 4);
        if (tid == 0) {
            float* o = out + it * 5;
            if (ok) { o[0] = bb.x; o[1] = bb.y; o[2] = bb.z; o[3] = bb.w; o[4] = rawsc[best]; }
            else    { o[0] = 0.f; o[1] = 0.f; o[2] = 0.f; o[3] = 0.f; o[4] = 0.f; }
        }
        const float ba = areas[best];
        for (int idx = tid; idx < NPROP; idx += 1024) {
            const float4 p  = *(const float4*)(props + idx * 4);
            const float xl = fmaxf(bb.x, p.x);
            const float yt = fmaxf(bb.y, p.y);
            const float xr = fminf(bb.z, p.z);
            const float yb = fminf(bb.w, p.w);
            const float inter = fmaxf(xr - xl, 0.f) * fmaxf(yb - yt, 0.f);
            const float iou   = inter / (ba + areas[idx] - inter);
            if (iou > 0.7f || idx == best) live[idx] = NEG_INF;
        }
        __syncthreads();
    }
}

// ---------------------------------------------------------------------------
extern "C" void kernel_launch(void* const* d_in, const int* in_sizes, int n_in,
                              void* d_out, int out_size, void* d_ws, size_t ws_size,
                              hipStream_t stream)
{
    (void)in_sizes; (void)n_in; (void)out_size; (void)ws_size;
    // setup_inputs order: image, feat, conv_w, conv_b, cls_w, cls_b, reg_w, reg_b
    const float* feat  = (const float*)d_in[1];   // (8,512,64,64); only batch 0 affects output
    const float* convw = (const float*)d_in[2];   // (512,512,3,3) OIHW
    const float* convb = (const float*)d_in[3];
    const float* clsw  = (const float*)d_in[4];   // (9,512)
    const float* clsb  = (const float*)d_in[5];
    const float* regw  = (const float*)d_in[6];   // (36,512)
    const float* regb  = (const float*)d_in[7];

    float* ws     = (float*)d_ws;
    float* t0     = ws;                       // 4096*512 = 8 MB
    float* props  = t0 + 4096 * C_IN;         // 36864*4
    float* rawsc  = props + NPROP * 4;        // 36864
    float* validf = rawsc + NPROP;            // 36864
    float* live   = validf + NPROP;           // 36864
    float* areas  = live + NPROP;             // 36864

    dim3 g1(32, 8);   // 32 M-tiles (128 each) x 8 N-tiles (64 each)
    rpn_conv3x3_wmma<<<g1, 256, 0, stream>>>(feat, convw, convb, t0);
    rpn_head_decode<<<(NPROP + 255) / 256, 256, 0, stream>>>(t0, clsw, clsb, regw, regb,
                                                             props, rawsc, validf);
    rpn_topk_nms<<<1, 1024, 0, stream>>>(props, rawsc, validf, live, areas, (float*)d_out);
}